// TokenPooler_16913581212147
// MI455X (gfx1250) — compile-verified
//
#include <hip/hip_runtime.h>
#include <hip/hip_bf16.h>
#include <math.h>

// Problem constants (from the reference)
#define BATCH    128
#define SEQ      1024
#define HID      768
#define MAXV     25
#define SLEVI    (SEQ - 2 * MAXV)   // 974

typedef __attribute__((ext_vector_type(2))) float v2f;
typedef __attribute__((ext_vector_type(8))) float v8f;

// -------------------------------------------------------------------------
// Kernel 1: masked attention pooling.
// One block per batch (256 threads = 8 wave32).  Only the <=48 active token
// rows of `hidden` are ever read (s in [974,974+v) U [999,999+v)).
// The subject/bias score terms are constant per batch and cancel in softmax.
// -------------------------------------------------------------------------
__global__ __launch_bounds__(256) void
tokenpool_scores_kernel(const float* __restrict__ hidden,
                        const long long* __restrict__ verb_count, // int64
                        const float* __restrict__ align_w,        // (2H,)
                        float* __restrict__ pooled)               // (B,H) ws
{
    const int b    = blockIdx.x;
    const int tid  = threadIdx.x;
    const int lane = tid & 31;
    const int wave = tid >> 5;

    __shared__ float s_score[2 * MAXV];
    __shared__ float s_wt[2 * MAXV];

    const int v  = (int)verb_count[b];
    const int nt = 2 * v;                       // number of active tokens
    const float* hb = hidden + (size_t)b * SEQ * HID;
    const float* w2 = align_w + HID;            // align_w[0, H:2H]

    // --- per-token scores: one wave per token, lane-strided dot over H ---
    for (int t = wave; t < nt; t += 8) {
        const int s = (t < v) ? (SLEVI + t) : (SLEVI + MAXV + (t - v));
        const float* row = hb + (size_t)s * HID;
        float acc = 0.0f;
#pragma unroll
        for (int h = lane; h < HID; h += 32)
            acc = fmaf(row[h], w2[h], acc);
        // wave32 butterfly reduction
        for (int off = 16; off > 0; off >>= 1)
            acc += __shfl_xor(acc, off, 32);
        if (lane == 0) s_score[t] = acc;
    }
    __syncthreads();

    // --- softmax over <=48 scores (serial in thread 0: trivial size) ---
    if (tid == 0) {
        float m = -INFINITY;
        for (int t = 0; t < nt; ++t) m = fmaxf(m, s_score[t]);
        if (!isfinite(m)) m = 0.0f;             // nt==0 case
        float denom = 0.0f;
        for (int t = 0; t < nt; ++t) {
            float e = expf(s_score[t] - m);
            s_wt[t] = e;
            denom += e;
        }
        const float inv = (denom > 0.0f) ? (1.0f / fmaxf(denom, 1e-30f)) : 0.0f;
        for (int t = 0; t < nt; ++t) s_wt[t] *= inv;
    }
    __syncthreads();

    // --- pooled[b,h] = sum_t w_t * hidden[b, s_t, h] ---
    for (int h = tid; h < HID; h += 256) {
        float acc = 0.0f;
        for (int t = 0; t < nt; ++t) {
            const int s = (t < v) ? (SLEVI + t) : (SLEVI + MAXV + (t - v));
            acc = fmaf(s_wt[t], hb[(size_t)s * HID + h], acc);
        }
        pooled[(size_t)b * HID + h] = acc;
    }
}

// -------------------------------------------------------------------------
// Kernel 2: out = tanh(pooled @ out_w.T + out_b) via V_WMMA_F32_16X16X4_F32.
//   M = 128 (8 tiles of 16), N = 768 (48 tiles), K = 768 (192 steps of 4).
// Full fp32 — exact wrt the fp32 reference GEMM.
// One 16x16 output tile per wave; 8 waves per block cover M; grid.x covers N.
//
// Fragment layouts (ISA 7.12.2, 32-bit, wave32):
//   A (16x4, MxK):  lanes 0-15 hold M=lane; VGPR0/1 = K=(0,1); lanes 16-31
//                   hold the same M with K=(2,3)  -> per-lane float2 at
//                   pooled[m*K + k0 + 2*(lane>=16)].
//   B (4x16, KxN):  rows (K) striped over VGPRs, N over lanes — since we need
//                   B[k][n] = out_w[n][k], each lane reads a contiguous
//                   float2 from row-major out_w: the transpose is free.
//   C/D (16x16):    VGPR r = row (r + 8*(lane>=16)), col = lane&15.
// -------------------------------------------------------------------------
__global__ __launch_bounds__(256) void
gemm_tanh_wmma_kernel(const float* __restrict__ pooled,  // (128, 768)
                      const float* __restrict__ out_w,   // (768, 768) row-major
                      const float* __restrict__ out_b,   // (768,)
                      float* __restrict__ out)           // (128, 768)
{
    const int lane  = threadIdx.x & 31;
    const int wave  = threadIdx.x >> 5;
    const int half  = lane >> 4;        // 0: K pair (0,1); 1: K pair (2,3)
    const int l16   = lane & 15;

    const int mtile = wave;             // 0..7  -> rows  [16*mtile, +16)
    const int ntile = blockIdx.x;       // 0..47 -> cols  [16*ntile, +16)

    const int m = mtile * 16 + l16;     // A-fragment row for this lane
    const int n = ntile * 16 + l16;     // B-fragment col for this lane

    const float* arow = pooled + (size_t)m * HID;
    const float* brow = out_w  + (size_t)n * HID;   // out_w[n, :] == B[:, n]

    v8f c = {};
#pragma unroll 4
    for (int k0 = 0; k0 < HID; k0 += 4) {
        const int kb = k0 + 2 * half;
        v2f a = *(const v2f*)(arow + kb);
        v2f bf = *(const v2f*)(brow + kb);
        // D = A * B + C   (fp32, exact)
        c = __builtin_amdgcn_wmma_f32_16x16x4_f32(
                /*neg_a=*/false, a, /*neg_b=*/false, bf,
                /*c_mod=*/(short)0, c, /*reuse_a=*/false, /*reuse_b=*/false);
    }

    const float bias = out_b[ntile * 16 + l16];
#pragma unroll
    for (int r = 0; r < 8; ++r) {
        const int row = mtile * 16 + r + 8 * half;
        out[(size_t)row * HID + ntile * 16 + l16] = tanhf(c[r] + bias);
    }
}

// -------------------------------------------------------------------------
// Launch.  Inputs (setup_inputs order):
//  0 hidden (B,S,H) f32 | 1 verb_count (B,) i64 | 2 subject_hidden (B,H) f32
//  3 align_w (1,2H) f32 | 4 align_b (1,) f32    | 5 out_w (H,H) f32
//  6 out_b (H,) f32
// subject_hidden/align_b cancel in the softmax -> unused.
// -------------------------------------------------------------------------
extern "C" void kernel_launch(void* const* d_in, const int* in_sizes, int n_in,
                              void* d_out, int out_size, void* d_ws, size_t ws_size,
                              hipStream_t stream) {
    const float*     hidden     = (const float*)d_in[0];
    const long long* verb_count = (const long long*)d_in[1];
    const float*     align_w    = (const float*)d_in[3];
    const float*     out_w      = (const float*)d_in[5];
    const float*     out_b      = (const float*)d_in[6];
    float*           out        = (float*)d_out;

    float* pooled = (float*)d_ws;       // (128, 768) fp32 scratch

    tokenpool_scores_kernel<<<BATCH, 256, 0, stream>>>(
        hidden, verb_count, align_w, pooled);

    gemm_tanh_wmma_kernel<<<HID / 16, 256, 0, stream>>>(
        pooled, out_w, out_b, out);
}